// RealtimeHazeGNN_48584670053101
// MI455X (gfx1250) — compile-verified
//
#include <hip/hip_runtime.h>
#include <hip/hip_bf16.h>
#include <math.h>

typedef float v2f __attribute__((ext_vector_type(2)));
typedef float v8f __attribute__((ext_vector_type(8)));
typedef int   v4i __attribute__((vector_size(4 * sizeof(int))));

#define HID   128
#define HEADS 4
#define CH    32
#define SLOPE 0.2f
#define LN_EPS 1e-5f

#define KC    32            // K panel staged in LDS per iteration
#define APAD  4             // A row pad (stride KC+4=36 -> conflict-free 16x4 fragment reads)
#define BPAD  8             // B row pad (stride N+8  -> half-wave bank windows disjoint)
#define MAXN  128

#if defined(__gfx1250__) && \
    __has_builtin(__builtin_amdgcn_global_load_async_to_lds_b128) && \
    __has_builtin(__builtin_amdgcn_s_wait_asynccnt)
#define HAS_ASYNC_LDS 1
#else
#define HAS_ASYNC_LDS 0
#endif

__device__ __forceinline__ void cp_g2l_b128(const float* g, float* l) {
#if HAS_ASYNC_LDS
  __builtin_amdgcn_global_load_async_to_lds_b128(
      (__attribute__((address_space(1))) v4i*)g,
      (__attribute__((address_space(3))) v4i*)l, 0, 0);
#else
  *(float4*)l = *(const float4*)g;
#endif
}

__device__ __forceinline__ void cp_g2l_commit() {
#if HAS_ASYNC_LDS
  __builtin_amdgcn_s_wait_asynccnt(0);
#endif
}

// ---------------------------------------------------------------------------
// WMMA f32 GEMM: C[M,N] = act(A[M,K] @ B[K,N] + bias), all row-major f32.
// grid.x = M/16 (M multiple of 16), blockDim = 256 (8 waves).
// Wave w owns column tile w (N/16 <= 8). K staged in KC-wide LDS panels via
// async global->LDS copies; V_WMMA_F32_16X16X4_F32 fed from ds_loads.
// ISA 7.12.2 layouts:
//   A 16x4:  lane L, m=L%16; v2f = A[m][k0+2*(L>=16) + {0,1}]
//   B 4x16:  lane L, n=L%16; {B[k0+2*(L>=16)][n], B[k0+2*(L>=16)+1][n]}
//   C 16x16: VGPR r, lane L: row = r + 8*(L>=16), col = L%16
// ---------------------------------------------------------------------------
template <int RELU>
__global__ __launch_bounds__(256)
void wmma_gemm_f32(const float* __restrict__ A, const float* __restrict__ B,
                   const float* __restrict__ bias, float* __restrict__ C,
                   int M, int N, int K) {
  __shared__ float As[16 * (KC + APAD)];       //  2.3 KB
  __shared__ float Bs[KC * (MAXN + BPAD)];     // 17.0 KB

  const int tid    = threadIdx.x;
  const int wave   = tid >> 5;
  const int lane   = tid & 31;
  const int lo     = lane & 15;
  const int hiHalf = lane >> 4;
  const int rowTile = blockIdx.x;
  const int nct    = N >> 4;                   // column tiles (<= 8)
  const bool active = (wave < nct);
  const int n      = (active ? wave : 0) * 16 + lo;
  const int NB     = N + BPAD;                 // padded LDS row stride for B

  v8f c = {0.f, 0.f, 0.f, 0.f, 0.f, 0.f, 0.f, 0.f};

  for (int kc = 0; kc < K; kc += KC) {
    // ---- stage B panel: KC x N (row stride NB in LDS), coalesced b128 ----
    const float* gB = B + (size_t)kc * N;
    const int bq = (KC * N) >> 2;                       // float4 chunks
    for (int i = tid; i < bq; i += 256) {
      const int idx = i << 2;
      const int kr  = idx / N;
      const int nc  = idx - kr * N;
      cp_g2l_b128(gB + (size_t)kr * N + nc, &Bs[kr * NB + nc]);
    }
    // ---- stage A tile: 16 x KC (row stride KC+APAD in LDS) ----
    const int aq = (16 * KC) >> 2;
    for (int i = tid; i < aq; i += 256) {
      const int r  = i >> 3;                            // KC/4 == 8 chunks/row
      const int cc = (i & 7) << 2;
      cp_g2l_b128(A + (size_t)(rowTile * 16 + r) * K + kc + cc,
                  &As[r * (KC + APAD) + cc]);
    }
    cp_g2l_commit();
    __syncthreads();

    if (active) {                                       // wave-uniform: EXEC stays all-1s
#pragma unroll
      for (int k0 = 0; k0 < KC; k0 += 4) {
        const int ka = k0 + 2 * hiHalf;
        v2f a = *(const v2f*)&As[lo * (KC + APAD) + ka];
        v2f b;
        b.x = Bs[ka * NB + n];
        b.y = Bs[(ka + 1) * NB + n];
        c = __builtin_amdgcn_wmma_f32_16x16x4_f32(false, a, false, b,
                                                  (short)0, c, false, false);
      }
    }
    __syncthreads();                                    // before next panel overwrite
  }

  if (active) {
    const float bv = bias ? bias[n] : 0.f;
#pragma unroll
    for (int r = 0; r < 8; ++r) {
      const int mr = rowTile * 16 + r + 8 * hiHalf;
      float v = c[r] + bv;
      if (RELU) v = fmaxf(v, 0.f);
      C[(size_t)mr * N + n] = v;
    }
  }
}

// ---------------------------------------------------------------------------
// Float atomic max via ordered-int trick (global_atomic_max/min). Init = -inf.
// ---------------------------------------------------------------------------
__device__ __forceinline__ void atomicMaxFloat(float* addr, float val) {
  if (val >= 0.f) atomicMax((int*)addr, __float_as_int(val));
  else            atomicMin((unsigned int*)addr, __float_as_uint(val));
}

__global__ void init_gat(float* __restrict__ acc, float* __restrict__ mbuf,
                         float* __restrict__ dbuf, int N) {
  int t = blockIdx.x * blockDim.x + threadIdx.x;
  if (t < N * HID) acc[t] = 0.f;
  if (t < N * HEADS) { mbuf[t] = __int_as_float(0xff800000); dbuf[t] = 0.f; }
}

__device__ __forceinline__ void edge_ids(int t, const int* __restrict__ src,
                                         const int* __restrict__ dst, int E,
                                         int& e, int& h, int& sv, int& dv) {
  e = t >> 2; h = t & 3;
  if (e < E) { sv = src[e]; dv = dst[e]; }
  else       { sv = dv = e - E; }          // appended self-loops
}

// Phase A: s[e,h] = att_h . leaky_relu(xl[src] + xr[dst]); segment max via atomics
__global__ void edge_logits(const float* __restrict__ xl, const float* __restrict__ xr,
                            const int* __restrict__ src, const int* __restrict__ dst,
                            const float* __restrict__ att,
                            float* __restrict__ s, float* __restrict__ mbuf,
                            int E, int N) {
  int t = blockIdx.x * blockDim.x + threadIdx.x;
  if (t >= (E + N) * HEADS) return;
  int e, h, sv, dv; edge_ids(t, src, dst, E, e, h, sv, dv);
  const float4* pl = (const float4*)(xl + (size_t)sv * HID + h * CH);
  const float4* pr = (const float4*)(xr + (size_t)dv * HID + h * CH);
  const float4* pa = (const float4*)(att + h * CH);
  float acc = 0.f;
#pragma unroll
  for (int c = 0; c < CH / 4; ++c) {
    float4 l = pl[c], r = pr[c], a = pa[c];
    float v0 = l.x + r.x, v1 = l.y + r.y, v2 = l.z + r.z, v3 = l.w + r.w;
    v0 = v0 > 0.f ? v0 : v0 * SLOPE;
    v1 = v1 > 0.f ? v1 : v1 * SLOPE;
    v2 = v2 > 0.f ? v2 : v2 * SLOPE;
    v3 = v3 > 0.f ? v3 : v3 * SLOPE;
    acc += a.x * v0 + a.y * v1 + a.z * v2 + a.w * v3;
  }
  s[t] = acc;
  atomicMaxFloat(&mbuf[(size_t)dv * HEADS + h], acc);
}

// Phase B: a = exp(s - m[dst]); denom[dst] += a (overwrite s with a)
__global__ void edge_exp(const int* __restrict__ src, const int* __restrict__ dst,
                         float* __restrict__ s, const float* __restrict__ mbuf,
                         float* __restrict__ dbuf, int E, int N) {
  int t = blockIdx.x * blockDim.x + threadIdx.x;
  if (t >= (E + N) * HEADS) return;
  int e, h, sv, dv; edge_ids(t, src, dst, E, e, h, sv, dv);
  float a = expf(s[t] - mbuf[(size_t)dv * HEADS + h]);
  s[t] = a;
  atomicAdd(&dbuf[(size_t)dv * HEADS + h], a);
}

// Phase C: out[dst] += (a/denom[dst]) * xl[src]
__global__ void edge_aggr(const int* __restrict__ src, const int* __restrict__ dst,
                          const float* __restrict__ xl, const float* __restrict__ s,
                          const float* __restrict__ dbuf, float* __restrict__ out,
                          int E, int N) {
  int t = blockIdx.x * blockDim.x + threadIdx.x;
  if (t >= (E + N) * HEADS) return;
  int e, h, sv, dv; edge_ids(t, src, dst, E, e, h, sv, dv);
  float alpha = s[t] / dbuf[(size_t)dv * HEADS + h];
  const float4* pl = (const float4*)(xl + (size_t)sv * HID + h * CH);
  float* po = out + (size_t)dv * HID + h * CH;
#pragma unroll
  for (int c = 0; c < CH / 4; ++c) {
    float4 l = pl[c];
    atomicAdd(&po[4 * c + 0], alpha * l.x);
    atomicAdd(&po[4 * c + 1], alpha * l.y);
    atomicAdd(&po[4 * c + 2], alpha * l.z);
    atomicAdd(&po[4 * c + 3], alpha * l.w);
  }
}

// Wave-per-node: out = elu( [resid +] LN(acc + bias; g, b) )
template <int RESID>
__global__ void node_post(const float* __restrict__ acc, const float* __restrict__ bias,
                          const float* __restrict__ g, const float* __restrict__ b,
                          const float* __restrict__ resid, float* __restrict__ out, int N) {
  const int wave = threadIdx.x >> 5;
  const int lane = threadIdx.x & 31;
  const int node = blockIdx.x * (blockDim.x >> 5) + wave;
  if (node >= N) return;
  const float* p = acc + (size_t)node * HID;
  float v[4], s = 0.f, ss = 0.f;
#pragma unroll
  for (int i = 0; i < 4; ++i) {
    int c = lane + 32 * i;
    float x = p[c] + bias[c];
    v[i] = x; s += x; ss += x * x;
  }
#pragma unroll
  for (int off = 16; off > 0; off >>= 1) {
    s  += __shfl_xor(s,  off, 32);
    ss += __shfl_xor(ss, off, 32);
  }
  const float mu   = s * (1.f / HID);
  const float var  = ss * (1.f / HID) - mu * mu;
  const float rstd = rsqrtf(var + LN_EPS);
  float* q = out + (size_t)node * HID;
#pragma unroll
  for (int i = 0; i < 4; ++i) {
    int c = lane + 32 * i;
    float y = (v[i] - mu) * rstd * g[c] + b[c];
    if (RESID) y += resid[(size_t)node * HID + c];
    q[c] = y > 0.f ? y : expm1f(y);        // ELU
  }
}

// Tiny-N head projection: out[M,Nout] = act(A[M,K] @ W[K,Nout] + bias)
template <int SOFTPLUS>
__global__ void head_out(const float* __restrict__ A, const float* __restrict__ W,
                         const float* __restrict__ bias, float* __restrict__ out,
                         int M, int K, int Nout) {
  int t = blockIdx.x * blockDim.x + threadIdx.x;
  if (t >= M * Nout) return;
  int m = t / Nout, n = t - m * Nout;
  float acc = bias[n];
  for (int k = 0; k < K; ++k) acc += A[(size_t)m * K + k] * W[(size_t)k * Nout + n];
  if (SOFTPLUS) acc = fmaxf(acc, 0.f) + log1pf(expf(-fabsf(acc)));  // stable softplus
  out[t] = acc;
}

// ---------------------------------------------------------------------------
extern "C" void kernel_launch(void* const* d_in, const int* in_sizes, int n_in,
                              void* d_out, int out_size, void* d_ws, size_t ws_size,
                              hipStream_t stream) {
  const float* x     = (const float*)d_in[0];
  const int*   ei    = (const int*)  d_in[1];
  const float* Wl1   = (const float*)d_in[2];
  const float* Wr1   = (const float*)d_in[3];
  const float* att1  = (const float*)d_in[4];
  const float* bias1 = (const float*)d_in[5];
  const float* g1    = (const float*)d_in[6];
  const float* b1    = (const float*)d_in[7];
  const float* Wl2   = (const float*)d_in[8];
  const float* Wr2   = (const float*)d_in[9];
  const float* att2  = (const float*)d_in[10];
  const float* bias2 = (const float*)d_in[11];
  const float* g2    = (const float*)d_in[12];
  const float* b2    = (const float*)d_in[13];
  const float* Wp1   = (const float*)d_in[14];
  const float* bp1   = (const float*)d_in[15];
  const float* Wp2   = (const float*)d_in[16];
  const float* bp2   = (const float*)d_in[17];
  const float* Wu1   = (const float*)d_in[18];
  const float* bu1   = (const float*)d_in[19];
  const float* Wu2   = (const float*)d_in[20];
  const float* bu2   = (const float*)d_in[21];

  const int N    = in_sizes[0] / 32;     // 50000 (multiple of 16)
  const int IN_  = 32;
  const int E    = in_sizes[1] / 2;      // 800000
  const int Etot = E + N;                // + self loops
  const int* src = ei;
  const int* dst = ei + E;

  // workspace layout (floats)
  float* ws   = (float*)d_ws;
  float* xl   = ws;                         // N*HID   (reused as t1[N*64])
  float* xr   = xl   + (size_t)N * HID;     // N*HID   (reused as u1[N*32])
  float* acc  = xr   + (size_t)N * HID;     // N*HID
  float* h1   = acc  + (size_t)N * HID;     // N*HID
  float* mbuf = h1   + (size_t)N * HID;     // N*HEADS
  float* dbuf = mbuf + (size_t)N * HEADS;   // N*HEADS
  float* sbuf = dbuf + (size_t)N * HEADS;   // Etot*HEADS

  float* pred = (float*)d_out;              // N*6
  float* unc  = pred + (size_t)N * 6;       // N*6

  const int BLK   = 256;
  const dim3 gGemm(N / 16);
  const int gInit = (N * HID + BLK - 1) / BLK;
  const int gEdge = (Etot * HEADS + BLK - 1) / BLK;
  const int gNode = (N + 7) / 8;            // 8 waves/block, 1 node/wave
  const int gHead = (N * 6 + BLK - 1) / BLK;

  // ---- layer 1 -----------------------------------------------------------
  wmma_gemm_f32<0><<<gGemm, BLK, 0, stream>>>(x, Wl1, nullptr, xl, N, HID, IN_);
  wmma_gemm_f32<0><<<gGemm, BLK, 0, stream>>>(x, Wr1, nullptr, xr, N, HID, IN_);
  init_gat<<<gInit, BLK, 0, stream>>>(acc, mbuf, dbuf, N);
  edge_logits<<<gEdge, BLK, 0, stream>>>(xl, xr, src, dst, att1, sbuf, mbuf, E, N);
  edge_exp   <<<gEdge, BLK, 0, stream>>>(src, dst, sbuf, mbuf, dbuf, E, N);
  edge_aggr  <<<gEdge, BLK, 0, stream>>>(src, dst, xl, sbuf, dbuf, acc, E, N);
  node_post<0><<<gNode, BLK, 0, stream>>>(acc, bias1, g1, b1, nullptr, h1, N);

  // ---- layer 2 (residual) ------------------------------------------------
  wmma_gemm_f32<0><<<gGemm, BLK, 0, stream>>>(h1, Wl2, nullptr, xl, N, HID, HID);
  wmma_gemm_f32<0><<<gGemm, BLK, 0, stream>>>(h1, Wr2, nullptr, xr, N, HID, HID);
  init_gat<<<gInit, BLK, 0, stream>>>(acc, mbuf, dbuf, N);
  edge_logits<<<gEdge, BLK, 0, stream>>>(xl, xr, src, dst, att2, sbuf, mbuf, E, N);
  edge_exp   <<<gEdge, BLK, 0, stream>>>(src, dst, sbuf, mbuf, dbuf, E, N);
  edge_aggr  <<<gEdge, BLK, 0, stream>>>(src, dst, xl, sbuf, dbuf, acc, E, N);
  node_post<1><<<gNode, BLK, 0, stream>>>(acc, bias2, g2, b2, h1, h1, N); // h1 = elu(h1 + LN(acc))

  // ---- heads -------------------------------------------------------------
  float* t1 = xl;   // N*64
  float* u1 = xr;   // N*32
  wmma_gemm_f32<1><<<gGemm, BLK, 0, stream>>>(h1, Wp1, bp1, t1, N, HID / 2, HID);
  head_out<0><<<gHead, BLK, 0, stream>>>(t1, Wp2, bp2, pred, N, HID / 2, 6);
  wmma_gemm_f32<1><<<gGemm, BLK, 0, stream>>>(h1, Wu1, bu1, u1, N, HID / 4, HID);
  head_out<1><<<gHead, BLK, 0, stream>>>(u1, Wu2, bu2, unc, N, HID / 4, 6);

  (void)n_in; (void)out_size; (void)ws_size;
}